// InteractionLayer_14216341750126
// MI455X (gfx1250) — compile-verified
//
#include <hip/hip_runtime.h>
#include <hip/hip_bf16.h>

typedef __attribute__((ext_vector_type(2))) float v2f;
typedef __attribute__((ext_vector_type(8))) float v8f;

#define NF 64          // features
#define DK 128         // depth
#define LDS_STRIDE 132 // 128 + 4 pad -> conflict-free ds_load_b64 fragment reads
#define NPAIRS 2016    // 64*63/2

__global__ __launch_bounds__(128)
void InteractionLayer_gram_triu_kernel(const float* __restrict__ x,
                                       float* __restrict__ out) {
    __shared__ float sx[NF * LDS_STRIDE]; // 33,792 bytes

    const int b   = blockIdx.x;
    const int tid = threadIdx.x;
    const float* __restrict__ xb = x + (size_t)b * (NF * DK);

    // ---- Stage x[b] (64x128 fp32) into padded LDS, float4 granularity ----
    // 64 rows * 32 float4 per row = 2048 float4 / 128 threads = 16 each.
    for (int idx = tid; idx < NF * (DK / 4); idx += 128) {
        const int row = idx >> 5;   // /32
        const int c4  = idx & 31;
        const float4 v = reinterpret_cast<const float4*>(xb)[(row << 5) + c4];
        float* dst = &sx[row * LDS_STRIDE + (c4 << 2)];
        dst[0] = v.x; dst[1] = v.y; dst[2] = v.z; dst[3] = v.w;
    }
    __syncthreads();

    const int wave  = tid >> 5;   // 0..3
    const int lane  = tid & 31;
    const int m     = lane & 15;  // A-frag: M row in tile; B-frag: N col in tile
    const int khalf = lane >> 4;  // selects K pair {0,1} vs {2,3} of the K=4 slab

    // wave -> 2x2 tile block of the 4x4 tile grid
    const int wm = (wave >> 1) << 1; // tile-row base: 0 or 2
    const int wn = (wave & 1)  << 1; // tile-col base: 0 or 2

    // Block (rows {2,3}, cols {0,1}) is entirely strictly-lower-triangular:
    // i >= 32, j <= 31 -> never j > i. Skip it.
    const bool dead = (wm == 2) && (wn == 0);

    if (!dead) {
        v8f acc[2][2];
        #pragma unroll
        for (int i = 0; i < 2; ++i)
            #pragma unroll
            for (int j = 0; j < 2; ++j)
                acc[i][j] = (v8f){0.f, 0.f, 0.f, 0.f, 0.f, 0.f, 0.f, 0.f};

        // ---- K loop: 32 steps of K=4; 4 frag loads + 4 WMMAs per step ----
        for (int k = 0; k < DK / 4; ++k) {
            const int kb = (k << 2) + (khalf << 1);
            v2f afrag[2], bfrag[2];
            #pragma unroll
            for (int i = 0; i < 2; ++i) {
                // A fragment for tile row (wm+i): lane m = M, holds A[M][kb..kb+1]
                afrag[i] = *(const v2f*)&sx[(((wm + i) << 4) + m) * LDS_STRIDE + kb];
                // B = A^T, so B-frag for tile col (wn+i): lane m = N, holds A[N][kb..kb+1]
                bfrag[i] = *(const v2f*)&sx[(((wn + i) << 4) + m) * LDS_STRIDE + kb];
            }
            #pragma unroll
            for (int i = 0; i < 2; ++i)
                #pragma unroll
                for (int j = 0; j < 2; ++j)
                    acc[i][j] = __builtin_amdgcn_wmma_f32_16x16x4_f32(
                        /*neg_a=*/false, afrag[i],
                        /*neg_b=*/false, bfrag[j],
                        /*c_mod=*/(short)0, acc[i][j],
                        /*reuse_a=*/false, /*reuse_b=*/false);
        }

        // ---- Scatter strictly-upper-triangular entries directly (packed) ----
        // D layout: VGPR v, lane L -> M = v + 8*(L/16), N = L%16.
        float* __restrict__ ob = out + (size_t)b * NPAIRS;
        #pragma unroll
        for (int i = 0; i < 2; ++i) {
            #pragma unroll
            for (int j = 0; j < 2; ++j) {
                const int ti = (wm + i) << 4;
                const int tj = (wn + j) << 4;
                #pragma unroll
                for (int v = 0; v < 8; ++v) {
                    const int gi = ti + v + (khalf << 3);
                    const int gj = tj + m;
                    if (gj > gi) {
                        // packed triu (k=1) row-major index
                        const int idx = gi * (NF - 1) - ((gi * (gi - 1)) >> 1) + (gj - gi - 1);
                        ob[idx] = acc[i][j][v];
                    }
                }
            }
        }
    }
}

extern "C" void kernel_launch(void* const* d_in, const int* in_sizes, int n_in,
                              void* d_out, int out_size, void* d_ws, size_t ws_size,
                              hipStream_t stream) {
    (void)in_sizes; (void)n_in; (void)d_ws; (void)ws_size; (void)out_size;
    const float* x = (const float*)d_in[0];
    float* out = (float*)d_out;
    const int batches = 4096;
    InteractionLayer_gram_triu_kernel<<<batches, 128, 0, stream>>>(x, out);
}